// CalibrationLayer_16853451669534
// MI455X (gfx1250) — compile-verified
//
#include <hip/hip_runtime.h>
#include <hip/hip_bf16.h>
#include <stdint.h>

// Empirical-CDF calibration (searchsorted + lerp), memory-bound.
// CDNA5 path: async global->LDS staging of the 4096-entry tables.

#define R_FIXED 4096
#define TPB 256
#define V4_PER_THREAD 2  // 8 elements per thread

typedef float v4f __attribute__((ext_vector_type(4)));

__device__ __forceinline__ float calib_one(float x,
                                           const float* __restrict__ sri,
                                           const float* __restrict__ sro,
                                           float ri0, float riL,
                                           float ro0, float roL) {
  // searchsorted(ri, x, side='right') for power-of-two R: count of elems <= x
  int pos = 0;
#pragma unroll
  for (int step = R_FIXED >> 1; step >= 1; step >>= 1) {
    int np = pos + step;
    pos = (sri[np - 1] <= x) ? np : pos;
  }
  int idx = pos;
  idx = idx < 1 ? 1 : idx;
  idx = idx > (R_FIXED - 1) ? (R_FIXED - 1) : idx;

  float x0 = sri[idx - 1];
  float x1 = sri[idx];
  float y0 = sro[idx - 1];
  float y1 = sro[idx];
  float yi = y0 + (y1 - y0) / (x1 - x0) * (x - x0);
  // reference: where(x>=ri[-1], ro[-1], where(x<=ri[0], ro[0], interp))
  yi = (x <= ri0) ? ro0 : yi;
  yi = (x >= riL) ? roL : yi;
  return yi;
}

__global__ __launch_bounds__(TPB) void calib_kernel_4096(
    const float* __restrict__ x, const float* __restrict__ ri,
    const float* __restrict__ ro, float* __restrict__ out, int n) {
  __shared__ __align__(16) float sri[R_FIXED];
  __shared__ __align__(16) float sro[R_FIXED];
  const int tid = threadIdx.x;

  // ---- Async global->LDS staging of both 16KB tables (CDNA5 async path) ----
  {
    uint64_t gri = (uint64_t)(uintptr_t)ri;
    uint64_t gro = (uint64_t)(uintptr_t)ro;
    // Low 32 bits of the flat shared-space pointer == wave-relative LDS byte
    // offset (ISA 10.2 aperture mapping).
    uint32_t lri = (uint32_t)(uintptr_t)&sri[0];
    uint32_t lro = (uint32_t)(uintptr_t)&sro[0];
#pragma unroll
    for (int r = 0; r < (R_FIXED * 4) / (TPB * 16); ++r) {  // 4 rounds x 4KB
      uint32_t off = (uint32_t)(r * TPB + tid) * 16u;
      asm volatile("global_load_async_to_lds_b128 %0, %1, %2"
                   :
                   : "v"(lri + off), "v"(off), "s"(gri)
                   : "memory");
      asm volatile("global_load_async_to_lds_b128 %0, %1, %2"
                   :
                   : "v"(lro + off), "v"(off), "s"(gro)
                   : "memory");
    }
    asm volatile("s_wait_asynccnt 0x0" ::: "memory");
  }
  __syncthreads();

  const float ri0 = sri[0];
  const float riL = sri[R_FIXED - 1];
  const float ro0 = sro[0];
  const float roL = sro[R_FIXED - 1];

  const v4f* __restrict__ xv = (const v4f*)x;
  v4f* __restrict__ ov = (v4f*)out;
  const int nv = n >> 2;

  long long vbase = (long long)blockIdx.x * (TPB * V4_PER_THREAD);
#pragma unroll
  for (int rr = 0; rr < V4_PER_THREAD; ++rr) {
    long long vi = vbase + (long long)rr * TPB + tid;
    if (vi < nv) {
      v4f xx = __builtin_nontemporal_load(xv + vi);  // streaming input
      v4f yy;
      yy.x = calib_one(xx.x, sri, sro, ri0, riL, ro0, roL);
      yy.y = calib_one(xx.y, sri, sro, ri0, riL, ro0, roL);
      yy.z = calib_one(xx.z, sri, sro, ri0, riL, ro0, roL);
      yy.w = calib_one(xx.w, sri, sro, ri0, riL, ro0, roL);
      __builtin_nontemporal_store(yy, ov + vi);  // streaming output
    }
  }

  // Scalar tail (n not a multiple of 4) -- empty for the reference shape.
  int tail_start = nv << 2;
  if (tail_start < n) {
    for (int i = tail_start + (int)blockIdx.x * TPB + tid; i < n;
         i += (int)gridDim.x * TPB) {
      out[i] = calib_one(x[i], sri, sro, ri0, riL, ro0, roL);
    }
  }
}

// Fallback for any table length (reads tables straight from global/L2).
__global__ __launch_bounds__(TPB) void calib_kernel_generic(
    const float* __restrict__ x, const float* __restrict__ ri,
    const float* __restrict__ ro, float* __restrict__ out, int n, int Rn) {
  int i = (int)blockIdx.x * TPB + (int)threadIdx.x;
  if (i >= n) return;
  float xv = x[i];
  int lo = 0, hi = Rn;  // first index with ri[idx] > x
  while (lo < hi) {
    int mid = (lo + hi) >> 1;
    if (ri[mid] <= xv) lo = mid + 1; else hi = mid;
  }
  int idx = lo;
  idx = idx < 1 ? 1 : idx;
  idx = idx > (Rn - 1) ? (Rn - 1) : idx;
  float x0 = ri[idx - 1], x1 = ri[idx];
  float y0 = ro[idx - 1], y1 = ro[idx];
  float yi = y0 + (y1 - y0) / (x1 - x0) * (xv - x0);
  yi = (xv <= ri[0]) ? ro[0] : yi;
  yi = (xv >= ri[Rn - 1]) ? ro[Rn - 1] : yi;
  out[i] = yi;
}

extern "C" void kernel_launch(void* const* d_in, const int* in_sizes, int n_in,
                              void* d_out, int out_size, void* d_ws,
                              size_t ws_size, hipStream_t stream) {
  (void)d_ws; (void)ws_size;
  const float* x  = (const float*)d_in[0];
  const float* ri = (const float*)d_in[1];
  const float* ro = (const float*)d_in[2];
  float* out = (float*)d_out;
  const int n = out_size;

  if (n_in >= 3 && in_sizes[1] == R_FIXED && in_sizes[2] == R_FIXED) {
    const int elems_per_block = TPB * 4 * V4_PER_THREAD;  // 2048
    int grid = (n + elems_per_block - 1) / elems_per_block;
    if (grid < 1) grid = 1;
    calib_kernel_4096<<<grid, TPB, 0, stream>>>(x, ri, ro, out, n);
  } else {
    int Rn = in_sizes[1];
    int grid = (n + TPB - 1) / TPB;
    if (grid < 1) grid = 1;
    calib_kernel_generic<<<grid, TPB, 0, stream>>>(x, ri, ro, out, n, Rn);
  }
}